// TGS_70342974374496
// MI455X (gfx1250) — compile-verified
//
#include <hip/hip_runtime.h>

typedef float v2f __attribute__((ext_vector_type(2)));
typedef float v8f __attribute__((ext_vector_type(8)));

#define N_NODES 50000
#define N_EDGES 800000
#define D_FEAT  64
#define KDIM    128   // 2*D_FEAT
#define HIDDEN  128

// Non-returning device-scope f32 atomic add: guaranteed GLOBAL_ATOMIC_ADD_F32
// (fire-and-forget RMW at L2, tracked by STOREcnt), no CAS fallback.
__device__ __forceinline__ void atomic_fadd_dev(float* addr, float v) {
    asm volatile("global_atomic_add_f32 %0, %1, off scope:SCOPE_DEV"
                 :: "v"(addr), "v"(v) : "memory");
}

// ---------------------------------------------------------------------------
// Kernel 1: edge gather + scatter-add.  One wave32 per edge row: 32 lanes each
// move a float2 (256B coalesced read of x[src]), then two f32 atomics into the
// L2-resident neigh accumulator.
// ---------------------------------------------------------------------------
__global__ __launch_bounds__(256) void TGS_scatter_kernel(
    const float* __restrict__ x,
    const long long* __restrict__ ei,   // [2, N_EDGES] int64
    float* __restrict__ neigh)          // [N_NODES, 64] f32 accumulator
{
    long long idx = (long long)blockIdx.x * blockDim.x + threadIdx.x;
    int e = (int)(idx >> 5);
    int p = (int)(idx & 31);
    if (e >= N_EDGES) return;
    int src = (int)ei[e];
    int dst = (int)ei[N_EDGES + e];
    const float2* xv = (const float2*)x;
    float2 v = xv[src * 32 + p];
    float* base = &neigh[dst * 64 + 2 * p];
    atomic_fadd_dev(base,     v.x);
    atomic_fadd_dev(base + 1, v.y);
}

// ---------------------------------------------------------------------------
// Kernel 2: fused concat + (h @ W^T + b) + ReLU using V_WMMA_F32_16X16X4_F32.
// Block = 256 threads = 8 waves; block computes a 16-row x 128-col output tile.
// h rows (x | neigh) staged in LDS with stride 132 (bank-conflict-free for the
// 16-row A-fragment reads). Wave w owns output columns [16w, 16w+16).
// ---------------------------------------------------------------------------
__global__ __launch_bounds__(256) void TGS_gemm_kernel(
    const float* __restrict__ x,      // [N_NODES, 64]
    const float* __restrict__ neigh,  // [N_NODES, 64]
    const float* __restrict__ W,      // [128, 128] row-major (out, in)
    const float* __restrict__ bias,   // [128]
    float* __restrict__ out)          // [N_NODES, 128]
{
    __shared__ float tile[16][132];   // 16 rows of h, padded stride

    const int row_base = blockIdx.x * 16;

    // Cooperative stage of h tile: 2048 floats, 8 per thread, coalesced.
    #pragma unroll
    for (int i = 0; i < 8; ++i) {
        int flat = threadIdx.x + 256 * i;
        int r = flat >> 7;          // 0..15
        int c = flat & 127;         // 0..127
        int node = row_base + r;
        float v = (c < 64) ? x[node * 64 + c] : neigh[node * 64 + (c - 64)];
        tile[r][c] = v;
    }
    __syncthreads();

    const int wave = threadIdx.x >> 5;   // wave32: 8 waves per block
    const int lane = threadIdx.x & 31;
    const int col_base = wave * 16;

    const int row  = lane & 15;          // A: M index / B: N index
    const int half = lane >> 4;          // 0: K={0,1}, 1: K={2,3}
    const int koff = half * 2;
    const int wc   = col_base + row;     // W row feeding this lane's B frag

    v8f acc = {};
    #pragma unroll 4
    for (int k0 = 0; k0 < KDIM; k0 += 4) {
        // A fragment: h[row][k0+koff .. +1]  (ds_load_2addr_b64, conflict-free)
        v2f a = *(const v2f*)&tile[row][k0 + koff];
        // B fragment: W^T[k][n] = W[wc][k0+koff .. +1]  (L2-resident)
        v2f bf = *(const v2f*)&W[wc * KDIM + k0 + koff];
        acc = __builtin_amdgcn_wmma_f32_16x16x4_f32(
            /*neg_a=*/false, a, /*neg_b=*/false, bf,
            /*c_mod=*/(short)0, acc, /*reuse_a=*/false, /*reuse_b=*/false);
    }

    // Epilogue: C/D layout — VGPR r: lanes 0-15 -> M=r, N=lane;
    //                                lanes 16-31 -> M=r+8, N=lane-16.
    const int n     = lane & 15;
    const int m_off = half * 8;
    const int gc    = col_base + n;
    const float bv  = bias[gc];
    #pragma unroll
    for (int r = 0; r < 8; ++r) {
        int gr = row_base + r + m_off;
        float v = acc[r] + bv;
        out[gr * HIDDEN + gc] = fmaxf(v, 0.0f);
    }
}

// ---------------------------------------------------------------------------
extern "C" void kernel_launch(void* const* d_in, const int* in_sizes, int n_in,
                              void* d_out, int out_size, void* d_ws, size_t ws_size,
                              hipStream_t stream) {
    const float*     x  = (const float*)d_in[0];
    const long long* ei = (const long long*)d_in[1];   // int64 edge_index
    const float*     W  = (const float*)d_in[2];
    const float*     b  = (const float*)d_in[3];
    float* out   = (float*)d_out;
    float* neigh = (float*)d_ws;                        // 50000*64 f32 = 12.8 MB

    // Zero the segment-sum accumulator (graph-capture-safe memset node).
    hipMemsetAsync(neigh, 0, (size_t)N_NODES * D_FEAT * sizeof(float), stream);

    // Scatter: E * 32 lanes of work.
    long long total = (long long)N_EDGES * 32;
    int blk = 256;
    int grid = (int)((total + blk - 1) / blk);
    TGS_scatter_kernel<<<grid, blk, 0, stream>>>(x, ei, neigh);

    // Fused concat + GEMM + bias + ReLU: 50000/16 = 3125 row tiles.
    TGS_gemm_kernel<<<N_NODES / 16, 256, 0, stream>>>(x, neigh, W, b, out);
}